// LearnableSparseTrigger_69793218560413
// MI455X (gfx1250) — compile-verified
//
#include <hip/hip_runtime.h>
#include <cstdint>
#include <math.h>

#define SEG      256
#define SMOOTH_K 9
#define HALF_K   (SMOOTH_K / 2)
#define BASE_AMP 0.08f
#define LLEN     4096
#define ROW      (2 * LLEN)        // 8192 floats = 32 KB per batch row
#define NTHREADS 256
#define CHUNKS   (ROW / 4 / NTHREADS)   // 8 x b128 per thread

#define GAS __attribute__((address_space(1)))
#define LAS __attribute__((address_space(3)))

typedef int v4i __attribute__((ext_vector_type(4)));

// ---------------- CDNA5 async global<->LDS helpers ----------------

__device__ __forceinline__ void async_load_b128(const float* gsrc, float* lds) {
#if __has_builtin(__builtin_amdgcn_global_load_async_to_lds_b128)
  __builtin_amdgcn_global_load_async_to_lds_b128(
      (GAS v4i*)(uintptr_t)gsrc,
      (LAS v4i*)(uintptr_t)lds, 0, 0);
#else
  unsigned int loff = (unsigned int)(uintptr_t)lds;
  unsigned long long ga = (unsigned long long)(uintptr_t)gsrc;
  asm volatile("global_load_async_to_lds_b128 %0, %1, off"
               :: "v"(loff), "v"(ga) : "memory");
#endif
}

__device__ __forceinline__ void async_store_b128(float* gdst, const float* lds) {
#if __has_builtin(__builtin_amdgcn_global_store_async_from_lds_b128)
  __builtin_amdgcn_global_store_async_from_lds_b128(
      (GAS v4i*)(uintptr_t)gdst,
      (LAS v4i*)(uintptr_t)lds, 0, 0);
#else
  unsigned int loff = (unsigned int)(uintptr_t)lds;
  unsigned long long ga = (unsigned long long)(uintptr_t)gdst;
  asm volatile("global_store_async_from_lds_b128 %0, %1, off"
               :: "v"(ga), "v"(loff) : "memory");
#endif
}

__device__ __forceinline__ void wait_async0() {
#if __has_builtin(__builtin_amdgcn_s_wait_asynccnt)
  __builtin_amdgcn_s_wait_asynccnt(0);
#else
  asm volatile("s_wait_asynccnt 0" ::: "memory");
#endif
}

// ---------------- kernel 1 (hot path): one block per batch row ----------------

__global__ void __launch_bounds__(NTHREADS)
trigger_main(const float* __restrict__ x, const float* __restrict__ pat,
             const int* __restrict__ starts, float* __restrict__ out) {
  __shared__ float row[ROW];                 // 32 KB staging buffer
  __shared__ float wpart[NTHREADS / 32];
  __shared__ float s_amp;

  const int b   = blockIdx.x;
  const int tid = threadIdx.x;
  const float* src = x   + (size_t)b * ROW;
  float*       dst = out + (size_t)b * ROW;

  // Stage the full row into LDS via the async DMA path (no VGPR traffic).
#pragma unroll
  for (int i = 0; i < CHUNKS; ++i) {
    int e = (i * NTHREADS + tid) * 4;
    async_load_b128(src + e, row + e);
  }
  wait_async0();
  __syncthreads();

  // Per-row sum of squares from LDS (ds_load_b128), wave32 + cross-wave reduce.
  float ss = 0.f;
#pragma unroll
  for (int i = 0; i < CHUNKS; ++i) {
    int e = (i * NTHREADS + tid) * 4;
    float4 v = *reinterpret_cast<const float4*>(row + e);
    ss = fmaf(v.x, v.x, ss);
    ss = fmaf(v.y, v.y, ss);
    ss = fmaf(v.z, v.z, ss);
    ss = fmaf(v.w, v.w, ss);
  }
#pragma unroll
  for (int o = 16; o > 0; o >>= 1) ss += __shfl_xor(ss, o, 32);
  if ((tid & 31) == 0) wpart[tid >> 5] = ss;
  __syncthreads();
  if (tid == 0) {
    float tsum = 0.f;
#pragma unroll
    for (int w = 0; w < NTHREADS / 32; ++w) tsum += wpart[w];
    s_amp = sqrtf(tsum * (1.0f / ROW) + 1e-12f);
  }
  __syncthreads();
  const float amp = s_amp;

  // Add the windowed update in LDS: 512 elements (both channels), tid covers one each.
  const int s = starts[b];
  row[s + tid]        += amp * pat[tid];
  row[LLEN + s + tid] += amp * pat[SEG + tid];
  __syncthreads();

  // Ship the updated row back to HBM via the async DMA path.
#pragma unroll
  for (int i = 0; i < CHUNKS; ++i) {
    int e = (i * NTHREADS + tid) * 4;
    async_store_b128(dst + e, row + e);
  }
  wait_async0();
}

// ---------------- block reduction (wave32) ----------------

__device__ __forceinline__ float block_sum_256(float v, float* wpart, int tid) {
#pragma unroll
  for (int o = 16; o > 0; o >>= 1) v += __shfl_xor(v, o, 32);
  if ((tid & 31) == 0) wpart[tid >> 5] = v;
  __syncthreads();
  float t = 0.f;
#pragma unroll
  for (int w = 0; w < NTHREADS / 32; ++w) t += wpart[w];
  __syncthreads();   // make wpart reusable
  return t;          // broadcast to every thread
}

// ---------------- kernel 2: effective_pattern -> d_ws[512] ----------------

__global__ void __launch_bounds__(SEG)
pattern_kernel(const float* __restrict__ pi, const float* __restrict__ pq,
               float* __restrict__ pat_out) {
  // Zero-padded by HALF_K on both sides -> branch-free 9-tap smoothing.
  __shared__ float p0[SEG + 2 * HALF_K], p1[SEG + 2 * HALF_K];
  __shared__ float wpart[SEG / 32];
  const int t = threadIdx.x;

  p0[HALF_K + t] = pi[t];
  p1[HALF_K + t] = pq[t];
  if (t < 2 * HALF_K) {                // zero the pads (8 entries each side total)
    int pad = (t < HALF_K) ? t : (SEG + HALF_K + (t - HALF_K));
    p0[pad] = 0.f;
    p1[pad] = 0.f;
  }
  __syncthreads();

  float a0 = 0.f, a1 = 0.f;
#pragma unroll
  for (int j = 0; j < SMOOTH_K; ++j) {
    a0 += p0[t + j];
    a1 += p1[t + j];
  }
  a0 *= (1.0f / SMOOTH_K);
  a1 *= (1.0f / SMOOTH_K);

  // per-channel demean
  float m0 = block_sum_256(a0, wpart, t) * (1.0f / SEG);
  float m1 = block_sum_256(a1, wpart, t) * (1.0f / SEG);
  float d0 = a0 - m0;
  float d1 = a1 - m1;

  // global RMS over both channels
  float ssq = block_sum_256(d0 * d0 + d1 * d1, wpart, t) * (1.0f / (2 * SEG));
  float scale = BASE_AMP * rsqrtf(ssq + 1e-8f);

  pat_out[t]       = d0 * scale;
  pat_out[SEG + t] = d1 * scale;
}

// ---------------- launch ----------------

extern "C" void kernel_launch(void* const* d_in, const int* in_sizes, int n_in,
                              void* d_out, int out_size, void* d_ws, size_t ws_size,
                              hipStream_t stream) {
  const float* x      = (const float*)d_in[0];
  const float* pi     = (const float*)d_in[1];
  const float* pq     = (const float*)d_in[2];
  const int*   starts = (const int*)d_in[3];
  float* pat = (float*)d_ws;                 // [2*SEG] scratch
  const int B = in_sizes[3];                 // 4096

  pattern_kernel<<<1, SEG, 0, stream>>>(pi, pq, pat);
  trigger_main<<<B, NTHREADS, 0, stream>>>(x, pat, starts, (float*)d_out);
}